// LUMLP_65687229826148
// MI455X (gfx1250) — compile-verified
//
#include <hip/hip_runtime.h>

// ---------------------------------------------------------------------------
// Problem constants (from reference setup_inputs)
// ---------------------------------------------------------------------------
#define BB   16
#define NN   2048
#define CC   256
#define RR   128
#define KNN  16
#define KG   8
#define NKG  (NN * KG)   // 16384
#define MT   4           // m-tiles per wave (64 rows)
#define NT   2           // n-tiles per wave (32 cols)

typedef __attribute__((ext_vector_type(2))) float v2f;
typedef __attribute__((ext_vector_type(8))) float v8f;

// CDNA5 fp32 WMMA: D(16x16) = A(16x4) * B(4x16) + C   [wave32, VOP3P]
__device__ __forceinline__ v8f wmma4(v2f a, v2f b, v8f c) {
  return __builtin_amdgcn_wmma_f32_16x16x4_f32(false, a, false, b, (short)0, c,
                                               false, false);
}

// ---------------------------------------------------------------------------
// 1) Brute-force kNN on xyz[:, :2, :], top-16 (ascending squared distance,
//    strict '<' insert while scanning m ascending => smaller index wins ties,
//    matching jax.lax.top_k(-dist, k)).
// ---------------------------------------------------------------------------
__global__ __launch_bounds__(128)
void knn16(const float* __restrict__ xyz, int* __restrict__ idx16) {
  __shared__ float sx[128], sy[128];
  const int b = blockIdx.y;
  const int n = blockIdx.x * 128 + threadIdx.x;
  const float* xb = xyz + (size_t)b * 3 * NN;       // channel 0
  const float* yb = xb + NN;                        // channel 1
  const float px = xb[n], py = yb[n];

  float dl[KNN]; int il[KNN];
#pragma unroll
  for (int j = 0; j < KNN; ++j) { dl[j] = 3.4e38f; il[j] = 0; }

  for (int mt = 0; mt < NN; mt += 128) {
    __syncthreads();
    sx[threadIdx.x] = xb[mt + threadIdx.x];
    sy[threadIdx.x] = yb[mt + threadIdx.x];
    __syncthreads();
    for (int t = 0; t < 128; ++t) {
      const float dx = px - sx[t], dy = py - sy[t];
      const float d = dx * dx + dy * dy;
      const int m = mt + t;
      if (d < dl[KNN - 1]) {
        // descending-j insertion: slot j reads old dl[j], old dl[j-1]
#pragma unroll
        for (int j = KNN - 1; j >= 1; --j) {
          const bool sh = d < dl[j - 1];
          const bool in = !sh && (d < dl[j]);
          dl[j] = sh ? dl[j - 1] : (in ? d : dl[j]);
          il[j] = sh ? il[j - 1] : (in ? m : il[j]);
        }
        if (d < dl[0]) { dl[0] = d; il[0] = m; }
      }
    }
  }
  int* o = idx16 + ((size_t)(b * NN + n)) * KNN;
#pragma unroll
  for (int j = 0; j < KNN; ++j) o[j] = il[j];
}

// ---------------------------------------------------------------------------
// 2) Generic batched GEMM: Out[b,m,n] = sum_k W[m,k]*X[b,k,n] + bias[m]
//    Compile-time K and column stride => every inner-loop load is a single
//    global_load with a literal IOFFSET. Each wave: 4 m-tiles x 2 n-tiles
//    (64x32 output, 64 acc VGPRs). A/B fragments live in distinct registers
//    (a[MT], bv[NT]) so all loads of a k-step issue as one clause with a
//    single s_wait, then 8 back-to-back WMMAs (8 independent chains).
//    A frag (16x4 f32): lane = {m = lane&15, kh = lane>>4};
//      vgpr0 = W[m][k+2*kh], vgpr1 = W[m][k+2*kh+1]
//    B frag (4x16 f32, mirrored). D frag: row = v + 8*kh, col = lane&15.
// ---------------------------------------------------------------------------
template <int K>
__global__ __launch_bounds__(128)
void gemm_wx(const float* __restrict__ W, const float* __restrict__ X,
             const float* __restrict__ bias, float* __restrict__ Out, int M) {
  const int b = blockIdx.z;
  const int m0 = blockIdx.y * (16 * MT);
  const int wave = threadIdx.x >> 5;
  const int lane = threadIdx.x & 31;
  const int n0 = (blockIdx.x * 4 + wave) * (16 * NT);
  const int ml = lane & 15;
  const int kh = lane >> 4;

  const float* Wp = W + (size_t)(m0 + ml) * K + 2 * kh;
  const float* Xp = X + ((size_t)b * K + 2 * kh) * NN + n0 + ml;

  v8f c[MT][NT] = {};
#pragma unroll 8
  for (int k = 0; k < K; k += 4) {
    if ((k & 31) == 0)   // compile-time after unroll; spec prefetch, no branch
      __builtin_prefetch(Xp + (size_t)(k + 32) * NN, 0, 1);
    v2f a[MT], bv[NT];
#pragma unroll
    for (int u = 0; u < NT; ++u) {
      bv[u].x = Xp[(size_t)k * NN + u * 16];
      bv[u].y = Xp[(size_t)(k + 1) * NN + u * 16];
    }
#pragma unroll
    for (int t = 0; t < MT; ++t)
      a[t] = *(const v2f*)(Wp + (size_t)(t * 16) * K + k);
#pragma unroll
    for (int t = 0; t < MT; ++t)
#pragma unroll
      for (int u = 0; u < NT; ++u)
        c[t][u] = wmma4(a[t], bv[u], c[t][u]);
  }

#pragma unroll
  for (int t = 0; t < MT; ++t) {
    const int mrow = m0 + t * 16 + 8 * kh;
#pragma unroll
    for (int u = 0; u < NT; ++u) {
      float* Op = Out + ((size_t)b * M + mrow) * NN + n0 + u * 16 + ml;
#pragma unroll
      for (int v = 0; v < 8; ++v)
        Op[(size_t)v * NN] = c[t][u][v] + bias[mrow + v];
    }
  }
}

// ---------------------------------------------------------------------------
// 3) BN(training) stats over [B, C, L] -> folded per-channel (a, b):
//    y = a*x + b with a = gamma*rsqrt(var+eps), b = beta - mean*a
// ---------------------------------------------------------------------------
__global__ __launch_bounds__(256)
void bn_stats(const float* __restrict__ X, const float* __restrict__ gamma,
              const float* __restrict__ beta, float2* __restrict__ ab,
              int C, int L) {
  const int c = blockIdx.x;
  const int t = threadIdx.x;
  float s = 0.f, s2 = 0.f;
  for (int bb = 0; bb < BB; ++bb) {
    const float* p = X + ((size_t)bb * C + c) * L;
    for (int i = t; i < L; i += 256) { const float v = p[i]; s += v; s2 += v * v; }
  }
  __shared__ float rs[256], rq[256];
  rs[t] = s; rq[t] = s2;
  __syncthreads();
  for (int off = 128; off; off >>= 1) {
    if (t < off) { rs[t] += rs[t + off]; rq[t] += rq[t + off]; }
    __syncthreads();
  }
  if (t == 0) {
    const float cnt = (float)BB * (float)L;
    const float mu = rs[0] / cnt;
    const float var = rq[0] / cnt - mu * mu;       // biased, like jnp.var
    const float a = gamma[c] * rsqrtf(var + 1e-5f);
    ab[c] = make_float2(a, beta[c] - mu * a);
  }
}

// Out[i] = relu(a*X[i]+b) (+ add[i]); channel = (i >> cshift) & cmask
__global__ __launch_bounds__(256)
void bn_apply(const float* __restrict__ X, const float2* __restrict__ ab,
              const float* __restrict__ add, float* __restrict__ Out,
              int cshift, int cmask, size_t total) {
  const size_t i = (size_t)blockIdx.x * 256 + threadIdx.x;
  if (i >= total) return;
  const float2 p = ab[(int)((i >> cshift) & (size_t)cmask)];
  float v = fmaxf(p.x * X[i] + p.y, 0.f);
  if (add) v += add[i];
  Out[i] = v;
}

__global__ void zero_f(float* p, int n) {
  const int i = blockIdx.x * 256 + threadIdx.x;
  if (i < n) p[i] = 0.f;
}

// ---------------------------------------------------------------------------
// 4) Group MLP GEMM over gathered columns, two passes (no 134MB h tensor):
//    column q in [0, N*KG): src = idx[kg*256 + (nn>>3)][nn&7], nn=q>>3, kg=q&7
//    h[:,q] = fc_w * x1bn[:, src] + fc_b
//    Each wave: 4 m-tiles x 2 gathered n-tiles; gather + strided B loads
//    amortized over 8 WMMAs.
//    MODE 0: accumulate per-row sum/sumsq (shfl + LDS + global f32 atomics)
//    MODE 1: recompute, y=relu(a*h+b), then max over the 8 q's of each point
//            (8 adjacent columns => cross-lane shfl_xor max over lane groups)
// ---------------------------------------------------------------------------
template <int MODE>
__global__ __launch_bounds__(128)
void group_gemm(const float* __restrict__ Wfc, const float* __restrict__ x1bn,
                const int* __restrict__ idx, const float* __restrict__ bias,
                float* __restrict__ gstats, const float2* __restrict__ ab,
                float* __restrict__ x2) {
  const int b = blockIdx.z;
  const int m0 = blockIdx.y * (16 * MT);
  const int wave = threadIdx.x >> 5;
  const int lane = threadIdx.x & 31;
  const int q0 = (blockIdx.x * 4 + wave) * (16 * NT);
  const int ml = lane & 15;
  const int kh = lane >> 4;

  // faithful torch .view remap: gat[b,r,kg*N+nn] = x[b,r, idx8[kg*256+nn/8][nn%8]]
  const float* XpU[NT];
#pragma unroll
  for (int u = 0; u < NT; ++u) {
    const int q = q0 + u * 16 + ml;
    const int nn = q >> 3, kg = q & 7;
    const int np = (kg << 8) + (nn >> 3), jp = nn & 7;
    const int s = idx[(((size_t)b << 11) + np) * KNN + jp];  // first 8 of top-16
    XpU[u] = x1bn + ((size_t)b * RR + 2 * kh) * NN + s;
  }
  const float* Wp = Wfc + (size_t)(m0 + ml) * RR + 2 * kh;

  v8f c[MT][NT] = {};
#pragma unroll 8
  for (int k = 0; k < RR; k += 4) {
    v2f a[MT], bv[NT];
#pragma unroll
    for (int u = 0; u < NT; ++u) {
      bv[u].x = XpU[u][(size_t)k * NN];       // literal-offset gathered loads
      bv[u].y = XpU[u][(size_t)(k + 1) * NN];
    }
#pragma unroll
    for (int t = 0; t < MT; ++t)
      a[t] = *(const v2f*)(Wp + (size_t)(t * 16) * RR + k);
#pragma unroll
    for (int t = 0; t < MT; ++t)
#pragma unroll
      for (int u = 0; u < NT; ++u)
        c[t][u] = wmma4(a[t], bv[u], c[t][u]);
  }

  if (MODE == 0) {
    __shared__ float srow[16 * MT * 2];  // [local row 0..63][sum,sumsq]
    srow[threadIdx.x] = 0.f;
    __syncthreads();
#pragma unroll
    for (int t = 0; t < MT; ++t) {
#pragma unroll
      for (int v = 0; v < 8; ++v) {
        const int lr = t * 16 + v + 8 * kh;        // local row 0..63
        float s1 = 0.f, sq = 0.f;
#pragma unroll
        for (int u = 0; u < NT; ++u) {
          const float val = c[t][u][v] + bias[m0 + lr];
          s1 += val; sq += val * val;
        }
#pragma unroll
        for (int m = 1; m <= 8; m <<= 1) {         // reduce 16 lanes/half-wave
          s1 += __shfl_xor(s1, m, 32);
          sq += __shfl_xor(sq, m, 32);
        }
        if (ml == 0) {
          atomicAdd(&srow[lr * 2 + 0], s1);
          atomicAdd(&srow[lr * 2 + 1], sq);
        }
      }
    }
    __syncthreads();
    atomicAdd(&gstats[m0 * 2 + threadIdx.x], srow[threadIdx.x]);
  } else {
#pragma unroll
    for (int t = 0; t < MT; ++t) {
#pragma unroll
      for (int u = 0; u < NT; ++u) {
#pragma unroll
        for (int v = 0; v < 8; ++v) {
          const int row = m0 + t * 16 + v + 8 * kh;
          const float2 p = ab[row];
          float val = fmaxf(p.x * (c[t][u][v] + bias[row]) + p.y, 0.f);
#pragma unroll
          for (int m = 1; m <= 4; m <<= 1)         // max over 8 adjacent cols
            val = fmaxf(val, __shfl_xor(val, m, 32));
          if ((ml & 7) == 0)
            x2[((size_t)b * RR + row) * NN + ((q0 + u * 16) >> 3) + (ml >> 3)] = val;
        }
      }
    }
  }
}

// gstats[2r]=sum, [2r+1]=sumsq over B*N*KG samples -> folded (a,b)
__global__ void group_finalize(const float* __restrict__ gstats,
                               const float* __restrict__ gamma,
                               const float* __restrict__ beta,
                               float2* __restrict__ ab) {
  const int r = threadIdx.x;
  const float cnt = (float)BB * (float)NKG;
  const float mu = gstats[2 * r] / cnt;
  const float var = gstats[2 * r + 1] / cnt - mu * mu;
  const float a = gamma[r] * rsqrtf(var + 1e-5f);
  ab[r] = make_float2(a, beta[r] - mu * a);
}

// ---------------------------------------------------------------------------
// 5) FeatureLaplacian with the faithful raw .view of gat2 [B,R,N*k]->[B,N,k,R]:
//    lin = n*2048 + j*128 + r ; r' = lin>>15 ; p' = lin&32767 indexes idx16 flat
//    lapT[b,r,n] = x2[b,r,n] - (1/16) * sum_j x2[b, r', idx16[b, p']]
// ---------------------------------------------------------------------------
__global__ __launch_bounds__(256)
void laplacian(const float* __restrict__ x2, const int* __restrict__ idx16,
               float* __restrict__ lapT) {
  const size_t tid = (size_t)blockIdx.x * 256 + threadIdx.x;  // b*R*N + r*N + n
  const int n = (int)(tid & (NN - 1));
  const int r = (int)((tid >> 11) & (RR - 1));
  const int b = (int)(tid >> 18);
  const float* xb = x2 + ((size_t)b << 18);       // b * 128 * 2048
  const int* ib = idx16 + ((size_t)b << 15);      // b * 2048 * 16
  float acc = 0.f;
  const int base = (n << 11) + r;
#pragma unroll
  for (int j = 0; j < KNN; ++j) {
    const int lin = base + (j << 7);
    const int rp = lin >> 15;
    const int pp = lin & 32767;                   // == flat [n''][j''] index
    acc += xb[(rp << 11) + ib[pp]];
  }
  lapT[tid] = xb[(r << 11) + n] - acc * 0.0625f;
}

// ---------------------------------------------------------------------------
// Host-side orchestration
// ---------------------------------------------------------------------------
extern "C" void kernel_launch(void* const* d_in, const int* in_sizes, int n_in,
                              void* d_out, int out_size, void* d_ws, size_t ws_size,
                              hipStream_t stream) {
  (void)in_sizes; (void)n_in; (void)out_size; (void)ws_size;
  const float* xyz   = (const float*)d_in[0];
  const float* feat  = (const float*)d_in[1];
  const float* w1    = (const float*)d_in[2];
  const float* b1    = (const float*)d_in[3];
  const float* bn1g  = (const float*)d_in[4];
  const float* bn1b  = (const float*)d_in[5];
  const float* fcw   = (const float*)d_in[6];
  const float* fcb   = (const float*)d_in[7];
  const float* bngg  = (const float*)d_in[8];
  const float* bngb  = (const float*)d_in[9];
  const float* luw   = (const float*)d_in[10];
  const float* lub   = (const float*)d_in[11];
  const float* bnlg  = (const float*)d_in[12];
  const float* bnlb  = (const float*)d_in[13];
  const float* w2    = (const float*)d_in[14];
  const float* b2    = (const float*)d_in[15];
  const float* bn2g  = (const float*)d_in[16];
  const float* bn2b  = (const float*)d_in[17];
  const float* w3    = (const float*)d_in[18];
  const float* b3    = (const float*)d_in[19];
  const float* bn3g  = (const float*)d_in[20];
  const float* bn3b  = (const float*)d_in[21];
  float* out = (float*)d_out;

  // workspace layout (256B aligned), with lifetime-safe reuse
  char* ws = (char*)d_ws;
  const size_t SZ_IDX = (size_t)BB * NN * KNN * sizeof(int);     // 2 MB
  const size_t SZ_RN  = (size_t)BB * RR * NN * sizeof(float);    // 16.78 MB
  const size_t SZ_CN  = (size_t)BB * CC * NN * sizeof(float);    // 33.55 MB
  size_t o = 0;
  auto alloc = [&](size_t bytes) { size_t r = o; o = (o + bytes + 255) & ~(size_t)255; return r; };
  int*    idx16p = (int*)   (ws + alloc(SZ_IDX));
  float*  x1raw  = (float*) (ws + alloc(SZ_RN));
  float*  x1bn   = (float*) (ws + alloc(SZ_RN));
  float*  x2     = (float*) (ws + alloc(SZ_RN));
  float*  lapT   = (float*) (ws + alloc(SZ_RN));
  float*  yraw   = (float*) (ws + alloc(SZ_CN));
  float*  yb     = (float*) (ws + alloc(SZ_CN));
  float2* ab1    = (float2*)(ws + alloc(RR * sizeof(float2)));
  float2* abg    = (float2*)(ws + alloc(RR * sizeof(float2)));
  float2* abl    = (float2*)(ws + alloc(RR * sizeof(float2)));
  float2* ab2    = (float2*)(ws + alloc(CC * sizeof(float2)));
  float2* ab3    = (float2*)(ws + alloc(2 * CC * sizeof(float2)));
  float*  gstats = (float*) (ws + alloc(2 * RR * sizeof(float)));
  float*  traw   = x1raw;   // x1raw dead after bn_apply -> x1bn
  float*  x3     = x1bn;    // x1bn dead after group_gemm pass 1

  const size_t totRN = (size_t)BB * RR * NN;
  const size_t totCN = (size_t)BB * CC * NN;
  const size_t tot2C = (size_t)BB * 2 * CC * NN;

  // 1) kNN (shared for KG=8 prefix and k=16)
  knn16<<<dim3(NN / 128, BB), 128, 0, stream>>>(xyz, idx16p);

  // 2) mlp1: x1raw = w1*feat + b1 ; BN+ReLU -> x1bn
  gemm_wx<CC><<<dim3(NN / 128, RR / 64, BB), 128, 0, stream>>>(w1, feat, b1, x1raw, RR);
  bn_stats<<<RR, 256, 0, stream>>>(x1raw, bn1g, bn1b, ab1, RR, NN);
  bn_apply<<<(int)((totRN + 255) / 256), 256, 0, stream>>>(x1raw, ab1, nullptr, x1bn, 11, RR - 1, totRN);

  // 3) DSgroupMLP: two-pass WMMA over gathered columns
  zero_f<<<1, 256, 0, stream>>>(gstats, 2 * RR);
  group_gemm<0><<<dim3(NKG / 128, RR / 64, BB), 128, 0, stream>>>(fcw, x1bn, idx16p, fcb, gstats, nullptr, nullptr);
  group_finalize<<<1, RR, 0, stream>>>(gstats, bngg, bngb, abg);
  group_gemm<1><<<dim3(NKG / 128, RR / 64, BB), 128, 0, stream>>>(fcw, x1bn, idx16p, fcb, nullptr, abg, x2);

  // 4) FeatureLaplacian + Linear + BN + residual
  laplacian<<<(int)(totRN / 256), 256, 0, stream>>>(x2, idx16p, lapT);
  gemm_wx<RR><<<dim3(NN / 128, RR / 64, BB), 128, 0, stream>>>(luw, lapT, lub, traw, RR);
  bn_stats<<<RR, 256, 0, stream>>>(traw, bnlg, bnlb, abl, RR, NN);
  bn_apply<<<(int)((totRN + 255) / 256), 256, 0, stream>>>(traw, abl, x2, x3, 11, RR - 1, totRN);

  // 5) mlp2 + residual feat
  gemm_wx<RR><<<dim3(NN / 128, CC / 64, BB), 128, 0, stream>>>(w2, x3, b2, yraw, CC);
  bn_stats<<<CC, 256, 0, stream>>>(yraw, bn2g, bn2b, ab2, CC, NN);
  bn_apply<<<(int)((totCN + 255) / 256), 256, 0, stream>>>(yraw, ab2, feat, yb, 11, CC - 1, totCN);

  // 6) mlp3 -> d_out raw, BN+ReLU in place
  gemm_wx<CC><<<dim3(NN / 128, (2 * CC) / 64, BB), 128, 0, stream>>>(w3, yb, b3, out, 2 * CC);
  bn_stats<<<2 * CC, 256, 0, stream>>>(out, bn3g, bn3b, ab3, 2 * CC, NN);
  bn_apply<<<(int)((tot2C + 255) / 256), 256, 0, stream>>>(out, ab3, nullptr, out, 11, 2 * CC - 1, tot2C);
}